// TopKLearnAttentionModule_47794396070416
// MI455X (gfx1250) — compile-verified
//
#include <hip/hip_runtime.h>
#include <math.h>

typedef float v2f __attribute__((ext_vector_type(2)));
typedef float v8f __attribute__((ext_vector_type(8)));

#define N_NODES 20000
#define DEG 33
#define HDIM 128

// workspace float offsets
#define OFF_QT   0
#define OFF_KE   (OFF_QT + 2048)
#define OFF_VT   (OFF_KE + 2048)
#define OFF_T1   (OFF_VT + 2048)
#define OFF_T2   (OFF_T1 + 2048)
#define OFF_T3   (OFF_T2 + 2048)   // stored 16x128, only 4 rows used
#define OFF_L    (OFF_T3 + 2048)   // 16x16
#define OFF_TGT  (OFF_L + 256)     // 16
#define OFF_KA   (OFF_TGT + 16)
#define OFF_KB   (OFF_KA + 2048)
#define OFF_C1   (OFF_KB + 2048)
#define OFF_C2   (OFF_C1 + 2048)
#define OFF_C3   (OFF_C2 + 2048)
#define OFF_R1   (OFF_C3 + 2048)

// ---------------------------------------------------------------------------
// 16x128 = (16x128 A, zero-padded past arows) @ (128x128 B), one wave, WMMA
// f32 16x16x4.  A frag: lanes 0-15 -> K={k0,k0+1}, lanes 16-31 -> K={k0+2,k0+3}.
// D: VGPR v -> row v (lanes 0-15) / row v+8 (lanes 16-31), col = lane%16.
// ---------------------------------------------------------------------------
__device__ __forceinline__ void gemm16(const float* __restrict__ A, int arows,
                                       const float* __restrict__ B,
                                       float* __restrict__ C, int lane) {
  v8f acc[8];
#pragma unroll
  for (int t = 0; t < 8; ++t) acc[t] = (v8f)0.0f;
  const int row   = lane & 15;
  const int khalf = (lane >> 4) << 1;        // 0 or 2
  const bool rowOK = row < arows;
  for (int k0 = 0; k0 < 128; k0 += 4) {
    const int kk = k0 + khalf;
    v2f a;
    a.x = rowOK ? A[row * 128 + kk]     : 0.0f;
    a.y = rowOK ? A[row * 128 + kk + 1] : 0.0f;
#pragma unroll
    for (int t = 0; t < 8; ++t) {
      const int col = t * 16 + row;          // B[k][col], col striped over lanes
      v2f b;
      b.x = B[kk * 128 + col];
      b.y = B[(kk + 1) * 128 + col];
      acc[t] = __builtin_amdgcn_wmma_f32_16x16x4_f32(
          false, a, false, b, (short)0, acc[t], false, false);
    }
  }
  const int col = lane & 15;
  const int rowAdd = (lane < 16) ? 0 : 8;
#pragma unroll
  for (int t = 0; t < 8; ++t)
#pragma unroll
    for (int v = 0; v < 8; ++v)
      C[(v + rowAdd) * 128 + t * 16 + col] = acc[t][v];
}

// ---------------------------------------------------------------------------
// Single-block precompute: all 16-row tables via WMMA.
// ---------------------------------------------------------------------------
__global__ void __launch_bounds__(256) tkla_precompute(
    const float* __restrict__ emb_virtual, const float* __restrict__ emb_reciever,
    const float* __restrict__ emb_edge,    const float* __restrict__ emb_static,
    const float* __restrict__ Wq,  const float* __restrict__ Wk,
    const float* __restrict__ Wv,  const float* __restrict__ Wek,
    const float* __restrict__ Wev, const float* __restrict__ Wcomb,
    const float* __restrict__ kp_w1, const float* __restrict__ kp_b1,
    const float* __restrict__ kp_w2, const float* __restrict__ kp_b2,
    float* __restrict__ ws) {
  const int lane = threadIdx.x & 31;
  const int wave = threadIdx.x >> 5;

  // ---- stage 1: independent 16x128 GEMMs, one per wave ----
  switch (wave) {
    case 0: gemm16(emb_virtual, 16, Wq,              ws + OFF_QT, lane); break;
    case 1: gemm16(emb_virtual, 16, Wk,              ws + OFF_KA, lane); break;
    case 2: gemm16(emb_reciever,16, Wek,             ws + OFF_KB, lane); break;
    case 3: gemm16(emb_virtual, 16, Wv,              ws + OFF_VT, lane); break;
    case 4: gemm16(emb_edge,    16, Wcomb,           ws + OFF_C1, lane); break;
    case 5: gemm16(emb_edge,    16, Wcomb + 128*128, ws + OFF_C2, lane); break;
    case 6: gemm16(emb_static,   4, Wcomb + 256*128, ws + OFF_C3, lane); break;
    case 7: gemm16(emb_virtual, 16, kp_w1,           ws + OFF_R1, lane); break;
  }
  __syncthreads();

  // ---- stage 2 ----
  if (wave == 0)      gemm16(ws + OFF_C1, 16, Wev, ws + OFF_T1, lane);
  else if (wave == 1) gemm16(ws + OFF_C2, 16, Wev, ws + OFF_T2, lane);
  else if (wave == 2) gemm16(ws + OFF_C3, 16, Wev, ws + OFF_T3, lane);
  else if (wave == 3) {
    for (int i = lane; i < 2048; i += 32)
      ws[OFF_KE + i] = ws[OFF_KA + i] + ws[OFF_KB + i];
  } else if (wave == 4) {
    if (lane < 16) {  // rho / interpolation target per node code
      float acc = 0.0f;
      for (int k = 0; k < 128; ++k) {
        float r = ws[OFF_R1 + lane * 128 + k] + kp_b1[k];
        r = fmaxf(r, 0.0f);
        acc = fmaf(r, kp_w2[k], acc);
      }
      acc += kp_b2[0];
      const float rho = 1.0f / (1.0f + expf(-acc));
      float kf = rho * (float)DEG;
      kf = fminf(fmaxf(kf, 1.0f), (float)DEG - 0.01f);
      ws[OFF_TGT + lane] = kf - 1.0f;
    }
  }
  __syncthreads();

  // ---- stage 3: L = QT @ KE^T / sqrt(128)  (16x16, K=128) ----
  if (wave == 0) {
    v8f acc = (v8f)0.0f;
    const int row   = lane & 15;     // doubles as B-col index
    const int khalf = (lane >> 4) << 1;
    const float* QT = ws + OFF_QT;
    const float* KE = ws + OFF_KE;
    for (int k0 = 0; k0 < 128; k0 += 4) {
      const int kk = k0 + khalf;
      v2f a; a.x = QT[row * 128 + kk]; a.y = QT[row * 128 + kk + 1];
      v2f b; b.x = KE[row * 128 + kk]; b.y = KE[row * 128 + kk + 1]; // B[k][c]=KE[c][k]
      acc = __builtin_amdgcn_wmma_f32_16x16x4_f32(
          false, a, false, b, (short)0, acc, false, false);
    }
    const float scale = 0.088388347648318447f;  // 1/sqrt(128)
    const int col = lane & 15;
    const int rowAdd = (lane < 16) ? 0 : 8;
#pragma unroll
    for (int v = 0; v < 8; ++v)
      ws[OFF_L + (v + rowAdd) * 16 + col] = acc[v] * scale;
  }
}

// ---------------------------------------------------------------------------
// Fused main pass: one wave per node; tables resident in LDS; streams
// node_out and the node's 33 edge_out rows (b128 stores).  Memory-bound.
// ---------------------------------------------------------------------------
__global__ void __launch_bounds__(256) tkla_fused(
    const int* __restrict__ node_states, const int* __restrict__ edge_states,
    const float* __restrict__ scalars,   const int* __restrict__ src_idx,
    const int* __restrict__ reverse_idx, const int* __restrict__ train_step,
    const float* __restrict__ emb_virtual, const float* __restrict__ emb_edge,
    const float* __restrict__ ws, float* __restrict__ out) {
  __shared__ float sVT[2048];
  __shared__ float sT1[2048];
  __shared__ float sT2[2048];
  __shared__ float sT3[512];
  __shared__ float sEMBV[2048];
  __shared__ float sEMBE[2048];
  __shared__ float sL[256];
  __shared__ float sTGT[16];
  __shared__ float sLogit[8][DEG];
  __shared__ float sSel[8][DEG];
  __shared__ int   sCode[8][DEG];

  const int t = threadIdx.x;
  for (int i = t; i < 2048; i += 256) {
    sVT[i]   = ws[OFF_VT + i];
    sT1[i]   = ws[OFF_T1 + i];
    sT2[i]   = ws[OFF_T2 + i];
    sEMBV[i] = emb_virtual[i];
    sEMBE[i] = emb_edge[i];
  }
  for (int i = t; i < 512; i += 256) sT3[i] = ws[OFF_T3 + i];
  if (t < 256) sL[t] = ws[OFF_L + t];
  if (t < 16)  sTGT[t] = ws[OFF_TGT + t];
  __syncthreads();

  const int lane = t & 31;
  const int w = t >> 5;
  const int n = blockIdx.x * 8 + w;   // 2500 blocks * 8 waves == 20000 exactly

  const int cn = 2 * (node_states[n * 3] + 2 * node_states[n * 3 + 1] +
                      4 * node_states[n * 3 + 2]);
  const float rec_s = scalars[(long)n * DEG];  // self-loop scalar (slot 0)

  // ---- per-edge codes + logits ----
  for (int d = lane; d < DEG; d += 32) {
    const long e = (long)n * DEG + d;
    const int src = src_idx[e];
    const int cs = 2 * (node_states[src * 3] + 2 * node_states[src * 3 + 1] +
                        4 * node_states[src * 3 + 2]);
    const int ec = edge_states[e * 4] + 2 * edge_states[e * 4 + 1] +
                   4 * edge_states[e * 4 + 2] + 8 * edge_states[e * 4 + 3];
    const long rv = reverse_idx[e];
    const int ecr = edge_states[rv * 4] + 2 * edge_states[rv * 4 + 1] +
                    4 * edge_states[rv * 4 + 2] + 8 * edge_states[rv * 4 + 3];
    const float se  = scalars[e];
    const float snd = scalars[(long)src * DEG];
    const int rlx  = (se < rec_s) ? 1 : 0;
    const int rlxd = ((snd + se) < rec_s) ? 1 : 0;
    const int sc = rlx + 2 * rlxd;
    sLogit[w][d] = sL[cn * 16 + cs];
    sCode[w][d]  = cs | (ec << 4) | (ecr << 8) | (sc << 12);
  }
  __syncthreads();

  // ---- interpolated k-th order statistic via rank counting (tie-safe) ----
  const float target = sTGT[cn];
  int idx_f = (int)floorf(target);
  idx_f = max(0, min(idx_f, DEG - 1));
  const int idx_c = min(idx_f + 1, DEG - 1);
  const float frac = target - (float)idx_f;

  float vf = -3.0e38f, vc = -3.0e38f;
  for (int d = lane; d < DEG; d += 32) {
    const float x = sLogit[w][d];
    int cgt = 0, cge = 0;
    for (int j = 0; j < DEG; ++j) {
      const float y = sLogit[w][j];
      cgt += (y > x) ? 1 : 0;
      cge += (y >= x) ? 1 : 0;
    }
    if (cgt <= idx_f && idx_f < cge) vf = x;  // x sits at desc. positions [cgt,cge)
    if (cgt <= idx_c && idx_c < cge) vc = x;
  }
  for (int off = 16; off > 0; off >>= 1) {
    vf = fmaxf(vf, __shfl_xor(vf, off, 32));
    vc = fmaxf(vc, __shfl_xor(vc, off, 32));
  }
  const float threshold = (1.0f - frac) * vf + frac * vc;

  // ---- straight-through annealed selection ----
  const int step = train_step[0];
  const bool eval = (step == -1);
  float tau = 0.05f;
  if (!eval) {
    const float ts = fminf((float)step / 10000.0f, 1.0f);
    tau = 4.0f + (0.1f - 4.0f) * ts;
  }
  for (int d = lane; d < DEG; d += 32) {
    const float diff = sLogit[w][d] - threshold;
    const float hard = (diff >= 0.0f) ? 1.0f : 0.0f;
    float sel;
    if (eval) {
      sel = hard;
    } else {
      const float soft = 1.0f / (1.0f + expf(-diff / tau));
      sel = (hard - soft) + soft;  // matches reference fp32 arithmetic
    }
    sSel[w][d] = sel;
  }
  __syncthreads();

  // ---- msg accumulation: each lane owns 4 channels ----
  const float4* sVT4   = (const float4*)sVT;
  const float4* sT14   = (const float4*)sT1;
  const float4* sT24   = (const float4*)sT2;
  const float4* sT34   = (const float4*)sT3;
  const float4* sEMBV4 = (const float4*)sEMBV;
  const float4* sEMBE4 = (const float4*)sEMBE;

  float4 acc = make_float4(0.f, 0.f, 0.f, 0.f);
  for (int d = 0; d < DEG; ++d) {
    const int pc = sCode[w][d];
    const float wd = sSel[w][d];
    const int cs = pc & 15, ec = (pc >> 4) & 15, ecr = (pc >> 8) & 15,
              sc = (pc >> 12) & 3;
    const float4 r0 = sVT4[cs * 32 + lane];
    const float4 r1 = sT14[ec * 32 + lane];
    const float4 r2 = sT24[ecr * 32 + lane];
    const float4 r3 = sT34[sc * 32 + lane];
    acc.x = fmaf(wd, r0.x + r1.x + r2.x + r3.x, acc.x);
    acc.y = fmaf(wd, r0.y + r1.y + r2.y + r3.y, acc.y);
    acc.z = fmaf(wd, r0.z + r1.z + r2.z + r3.z, acc.z);
    acc.w = fmaf(wd, r0.w + r1.w + r2.w + r3.w, acc.w);
  }

  // ---- node_out ----
  const float4 nf = sEMBV4[cn * 32 + lane];
  float4* nout = (float4*)out;
  nout[(long)n * 32 + lane] =
      make_float4(nf.x + acc.x, nf.y + acc.y, nf.z + acc.z, nf.w + acc.w);

  // ---- edge_out (33 rows for this node) ----
  float4* eout = (float4*)(out + (long)N_NODES * HDIM);
  for (int d = 0; d < DEG; ++d) {
    const int ec = (sCode[w][d] >> 4) & 15;
    const float4 ef = sEMBE4[ec * 32 + lane];
    eout[((long)n * DEG + d) * 32 + lane] =
        make_float4(ef.x + acc.x, ef.y + acc.y, ef.z + acc.z, ef.w + acc.w);
  }
}

// ---------------------------------------------------------------------------
extern "C" void kernel_launch(void* const* d_in, const int* in_sizes, int n_in,
                              void* d_out, int out_size, void* d_ws,
                              size_t ws_size, hipStream_t stream) {
  const int*   node_states  = (const int*)d_in[0];
  const int*   edge_states  = (const int*)d_in[1];
  const float* scalars      = (const float*)d_in[2];
  const int*   src_idx      = (const int*)d_in[3];
  // d_in[4] = dst_idx (implied by fixed-degree layout; unused)
  const int*   reverse_idx  = (const int*)d_in[5];
  const int*   train_step   = (const int*)d_in[6];
  const float* emb_virtual  = (const float*)d_in[7];
  const float* emb_reciever = (const float*)d_in[8];
  const float* emb_edge     = (const float*)d_in[9];
  const float* emb_static   = (const float*)d_in[10];
  const float* Wq    = (const float*)d_in[11];
  const float* Wk    = (const float*)d_in[12];
  const float* Wv    = (const float*)d_in[13];
  const float* Wek   = (const float*)d_in[14];
  const float* Wev   = (const float*)d_in[15];
  const float* Wcomb = (const float*)d_in[16];
  const float* kp_w1 = (const float*)d_in[17];
  const float* kp_b1 = (const float*)d_in[18];
  const float* kp_w2 = (const float*)d_in[19];
  const float* kp_b2 = (const float*)d_in[20];
  float* ws  = (float*)d_ws;
  float* out = (float*)d_out;

  hipLaunchKernelGGL(tkla_precompute, dim3(1), dim3(256), 0, stream,
                     emb_virtual, emb_reciever, emb_edge, emb_static,
                     Wq, Wk, Wv, Wek, Wev, Wcomb, kp_w1, kp_b1, kp_w2, kp_b2,
                     ws);
  hipLaunchKernelGGL(tkla_fused, dim3(N_NODES / 8), dim3(256), 0, stream,
                     node_states, edge_states, scalars, src_idx, reverse_idx,
                     train_step, emb_virtual, emb_edge, ws, out);
}